// GATP_43748536877305
// MI455X (gfx1250) — compile-verified
//
#include <hip/hip_runtime.h>
#include <hip/hip_bf16.h>

#define HID 128
#define NEG_SLOPE 0.2f
#define GAT_EPS 1e-16f

typedef float v2f __attribute__((ext_vector_type(2)));
typedef float v8f __attribute__((ext_vector_type(8)));

// ---------- monotone float<->uint mapping for atomic max on floats ----------
__device__ __forceinline__ unsigned flipf(float f) {
    unsigned u = __float_as_uint(f);
    return (u & 0x80000000u) ? ~u : (u | 0x80000000u);
}
__device__ __forceinline__ float unflipf(unsigned u) {
    return __uint_as_float((u & 0x80000000u) ? (u ^ 0x80000000u) : ~u);
}

// ---------- init kernels ----------
__global__ void init_node_kernel(unsigned* __restrict__ emax,
                                 float* __restrict__ denom, int n) {
    int i = blockIdx.x * blockDim.x + threadIdx.x;
    if (i < n) { emax[i] = 0u; denom[i] = 0.0f; }
}

__global__ void init_out_kernel(float* __restrict__ out,
                                const float* __restrict__ bias, int total) {
    int i = blockIdx.x * blockDim.x + threadIdx.x;
    if (i < total) out[i] = bias[i & (HID - 1)];
}

// ---------- WMMA GEMM: x = h @ W  (h: [n,128] f32, W: [128,128] f32) ----------
// One block = 16 rows of h; 8 waves, one 16x16 N-tile per wave; K in steps of 4.
__global__ __launch_bounds__(256) void gemm_x_kernel(const float* __restrict__ h,
                                                     const float* __restrict__ W,
                                                     float* __restrict__ x,
                                                     int n) {
    __shared__ float sh[16 * HID];  // 8 KB stripe of h
    const int block_row = blockIdx.x * 16;
    const int tid  = threadIdx.x;
    const int wave = tid >> 5;        // 0..7 -> N tile
    const int lane = tid & 31;

    // stage 16x128 stripe (2048 floats) with float4 loads
    const float4* gsrc = (const float4*)(h + (size_t)block_row * HID);
    float4*       ldst = (float4*)sh;
    #pragma unroll
    for (int i = tid; i < 16 * HID / 4; i += 256) ldst[i] = gsrc[i];
    __syncthreads();

    const int m    = lane & 15;            // A row within tile
    const int koff = (lane >> 4) << 1;     // lanes 16-31 hold K+2,K+3
    const int ncol = wave * 16 + (lane & 15);

    v8f c = {};
    #pragma unroll
    for (int k = 0; k < HID; k += 4) {
        v2f a, b;
        // A frag (16x4): lane m holds K={k+koff, k+koff+1}
        a.x = sh[m * HID + k + koff];
        a.y = sh[m * HID + k + koff + 1];
        // B frag (4x16): lane holds column ncol, K={k+koff, k+koff+1}
        b.x = W[(size_t)(k + koff)     * HID + ncol];
        b.y = W[(size_t)(k + koff + 1) * HID + ncol];
        c = __builtin_amdgcn_wmma_f32_16x16x4_f32(
                /*neg_a=*/false, a, /*neg_b=*/false, b,
                /*c_mod=*/(short)0, c, /*reuse_a=*/false, /*reuse_b=*/false);
    }

    // D layout: VGPR r -> M=r (lanes 0-15) / M=r+8 (lanes 16-31), N=lane&15
    const int rbase = block_row + ((lane >> 4) << 3);
    #pragma unroll
    for (int r = 0; r < 8; ++r) {
        int row = rbase + r;
        if (row < n) x[(size_t)row * HID + ncol] = c[r];
    }
}

// ---------- per-node attention dots: alpha_s = x.a_s, alpha_d = x.a_d ----------
__global__ __launch_bounds__(256) void attn_kernel(const float* __restrict__ x,
                                                   const float* __restrict__ a_s,
                                                   const float* __restrict__ a_d,
                                                   float* __restrict__ as_out,
                                                   float* __restrict__ ad_out,
                                                   int n) {
    int node = (blockIdx.x * blockDim.x + threadIdx.x) >> 5;
    int lane = threadIdx.x & 31;
    if (node >= n) return;
    float4 v  = ((const float4*)(x + (size_t)node * HID))[lane];
    float4 s4 = ((const float4*)a_s)[lane];
    float4 d4 = ((const float4*)a_d)[lane];
    float ps = v.x * s4.x + v.y * s4.y + v.z * s4.z + v.w * s4.w;
    float pd = v.x * d4.x + v.y * d4.y + v.z * d4.z + v.w * d4.w;
    #pragma unroll
    for (int o = 16; o > 0; o >>= 1) {
        ps += __shfl_down(ps, o, 32);
        pd += __shfl_down(pd, o, 32);
    }
    if (lane == 0) { as_out[node] = ps; ad_out[node] = pd; }
}

// ---------- pass 1: leaky-relu logits + segment max ----------
__global__ void edge_max_kernel(const int* __restrict__ src,
                                const int* __restrict__ dst,
                                const float* __restrict__ as_,
                                const float* __restrict__ ad_,
                                float* __restrict__ eval,
                                unsigned* __restrict__ emax,
                                int n_edges, int n_nodes) {
    int e = blockIdx.x * blockDim.x + threadIdx.x;
    int total = n_edges + n_nodes;
    if (e >= total) return;
    int s, d;
    if (e < n_edges) { s = src[e]; d = dst[e]; } else { s = d = e - n_edges; }
    float v = as_[s] + ad_[d];
    v = (v > 0.0f) ? v : NEG_SLOPE * v;
    eval[e] = v;
    atomicMax(emax + d, flipf(v));
}

// ---------- pass 2: exp + segment sum ----------
__global__ void edge_exp_kernel(const int* __restrict__ dst,
                                const float* __restrict__ eval,
                                const unsigned* __restrict__ emax,
                                float* __restrict__ exv,
                                float* __restrict__ denom,
                                int n_edges, int n_nodes) {
    int e = blockIdx.x * blockDim.x + threadIdx.x;
    int total = n_edges + n_nodes;
    if (e >= total) return;
    int d = (e < n_edges) ? dst[e] : (e - n_edges);
    float m  = unflipf(emax[d]);
    float ex = __expf(eval[e] - m);
    exv[e] = ex;
    atomicAdd(denom + d, ex);
}

// ---------- pass 3: weighted scatter aggregate (one wave per edge) ----------
__global__ __launch_bounds__(256) void edge_agg_kernel(const int* __restrict__ src,
                                                       const int* __restrict__ dst,
                                                       const float* __restrict__ x,
                                                       const float* __restrict__ exv,
                                                       const float* __restrict__ denom,
                                                       float* __restrict__ out,
                                                       int n_edges, int n_nodes) {
    long long gid = (long long)blockIdx.x * blockDim.x + threadIdx.x;
    int e    = (int)(gid >> 5);
    int lane = threadIdx.x & 31;
    int total = n_edges + n_nodes;
    if (e >= total) return;
    int s, d;
    if (e < n_edges) { s = src[e]; d = dst[e]; } else { s = d = e - n_edges; }
    float alpha = exv[e] / (denom[d] + GAT_EPS);
    float4 v = ((const float4*)(x + (size_t)s * HID))[lane];
    float* o = out + (size_t)d * HID + lane * 4;
    atomicAdd(o + 0, alpha * v.x);
    atomicAdd(o + 1, alpha * v.y);
    atomicAdd(o + 2, alpha * v.z);
    atomicAdd(o + 3, alpha * v.w);
}

// ---------- ReLU ----------
__global__ void relu_kernel(float* __restrict__ a, int total) {
    int i = blockIdx.x * blockDim.x + threadIdx.x;
    if (i < total) a[i] = fmaxf(a[i], 0.0f);
}

extern "C" void kernel_launch(void* const* d_in, const int* in_sizes, int n_in,
                              void* d_out, int out_size, void* d_ws, size_t ws_size,
                              hipStream_t stream) {
    const float* z     = (const float*)d_in[0];
    const int*   ei    = (const int*)  d_in[1];   // [2, E]
    const float* Ws    = (const float*)d_in[2];   // [3, 128, 128]
    const float* a_src = (const float*)d_in[3];   // [3, 128]
    const float* a_dst = (const float*)d_in[4];   // [3, 128]
    const float* bias  = (const float*)d_in[5];   // [3, 128]
    float*       out   = (float*)d_out;

    const int N = in_sizes[0] / HID;
    const int E = in_sizes[1] / 2;
    const int ET = E + N;                         // edges + self loops
    const int* src = ei;
    const int* dst = ei + E;

    // carve workspace (256B aligned)
    auto align = [](size_t v) { return (v + 255) & ~(size_t)255; };
    char* p = (char*)d_ws;
    float*    x      = (float*)p;    p += align((size_t)N * HID * sizeof(float));
    float*    bufA   = (float*)p;    p += align((size_t)N * HID * sizeof(float));
    float*    bufB   = (float*)p;    p += align((size_t)N * HID * sizeof(float));
    float*    as_    = (float*)p;    p += align((size_t)N * sizeof(float));
    float*    ad_    = (float*)p;    p += align((size_t)N * sizeof(float));
    unsigned* emax   = (unsigned*)p; p += align((size_t)N * sizeof(unsigned));
    float*    denom  = (float*)p;    p += align((size_t)N * sizeof(float));
    float*    eval   = (float*)p;    p += align((size_t)ET * sizeof(float));
    float*    exv    = (float*)p;    p += align((size_t)ET * sizeof(float));

    const float* h_ptr = z;
    for (int l = 0; l < 3; ++l) {
        float* o_ptr = (l == 0) ? bufA : (l == 1) ? bufB : out;
        const float* W  = Ws    + (size_t)l * HID * HID;
        const float* as = a_src + (size_t)l * HID;
        const float* ad = a_dst + (size_t)l * HID;
        const float* b  = bias  + (size_t)l * HID;

        init_node_kernel<<<(N + 255) / 256, 256, 0, stream>>>(emax, denom, N);
        init_out_kernel<<<((size_t)N * HID + 255) / 256, 256, 0, stream>>>(o_ptr, b, N * HID);

        gemm_x_kernel<<<(N + 15) / 16, 256, 0, stream>>>(h_ptr, W, x, N);

        attn_kernel<<<((size_t)N * 32 + 255) / 256, 256, 0, stream>>>(x, as, ad, as_, ad_, N);

        edge_max_kernel<<<(ET + 255) / 256, 256, 0, stream>>>(src, dst, as_, ad_, eval, emax, E, N);
        edge_exp_kernel<<<(ET + 255) / 256, 256, 0, stream>>>(dst, eval, emax, exv, denom, E, N);
        edge_agg_kernel<<<((size_t)ET * 32 + 255) / 256, 256, 0, stream>>>(src, dst, x, exv, denom,
                                                                           o_ptr, E, N);
        if (l < 2)
            relu_kernel<<<((size_t)N * HID + 255) / 256, 256, 0, stream>>>(o_ptr, N * HID);
        h_ptr = o_ptr;
    }
}